// JCIG_GNN_83004537962757
// MI455X (gfx1250) — compile-verified
//
#include <hip/hip_runtime.h>

typedef float v2f __attribute__((ext_vector_type(2)));
typedef float v8f __attribute__((ext_vector_type(8)));

#define N_NODES 50000
#define N_EDGES 800000
#define N_GRAPH 64
#define IN_DIM  384
#define HID     128
#define MLP_HID 64
#define STATS   3
#define FEAT    (3*HID + STATS)   // 387

// ---------------- degree / norm ----------------
__global__ void k_fill1(float* __restrict__ deg) {
  int i = blockIdx.x * blockDim.x + threadIdx.x;
  if (i < N_NODES) deg[i] = 1.0f;               // +1 self loop
}

__global__ void k_deg(const int* __restrict__ ei, float* __restrict__ deg) {
  int e = blockIdx.x * blockDim.x + threadIdx.x;
  if (e < N_EDGES) atomicAdd(&deg[ei[N_EDGES + e]], 1.0f);
}

__global__ void k_rsqrt(float* __restrict__ deg) {
  int i = blockIdx.x * blockDim.x + threadIdx.x;
  if (i < N_NODES) deg[i] = rsqrtf(deg[i]);     // deg -> dis
}

// ---------------- fp32 WMMA GEMM: H[N,128] = X[N,K] @ W[K,128] ----------------
// One wave = one 16x16 tile. Block = 8 waves = one M-tile x full 128 cols.
// N = 50000 = 3125 * 16 exactly; K % 4 == 0; no divergence (EXEC all ones).
__global__ __launch_bounds__(256) void k_gemm_wmma(const float* __restrict__ X,
                                                   const float* __restrict__ W,
                                                   float* __restrict__ H, int K) {
  const int lane  = threadIdx.x & 31;
  const int tileN = threadIdx.x >> 5;      // 0..7
  const int tileM = blockIdx.x;            // 0..3124
  const int half  = lane >> 4;             // 0/1 -> K offset +2
  const int l16   = lane & 15;
  const int row   = tileM * 16 + l16;      // A-matrix row held by this lane
  const int col   = tileN * 16 + l16;      // B/D column held by this lane

  v8f acc = {0.f, 0.f, 0.f, 0.f, 0.f, 0.f, 0.f, 0.f};
  const float* xa = X + (size_t)row * K + half * 2;        // a.x=K+0, a.y=K+1
  const float* wb = W + col + (size_t)(half * 2) * HID;    // b.x=row K, b.y=row K+1

  #pragma unroll 8
  for (int k = 0; k < K; k += 4) {
    float2 av = *(const float2*)(xa + k);
    v2f a; a.x = av.x; a.y = av.y;
    v2f b; b.x = wb[(size_t)k * HID];
           b.y = wb[(size_t)k * HID + HID];
    acc = __builtin_amdgcn_wmma_f32_16x16x4_f32(false, a, false, b,
                                                (short)0, acc, false, false);
  }

  // D layout: VGPR v holds row (v + 8*half), col = l16
  float* out = H + (size_t)(tileM * 16 + half * 8) * HID + tileN * 16 + l16;
  #pragma unroll
  for (int v = 0; v < 8; ++v) out[(size_t)v * HID] = acc[v];
}

// ---------------- self-loop init: agg = h * dis^2 ----------------
__global__ void k_selfinit(const float* __restrict__ h, const float* __restrict__ dis,
                           float* __restrict__ agg) {
  size_t i = (size_t)blockIdx.x * blockDim.x + threadIdx.x;   // N*HID threads
  int n = (int)(i >> 7);
  float d = dis[n];
  agg[i] = h[i] * d * d;
}

// ---------------- edge scatter: one wave per edge, 4 floats/lane ----------------
__global__ __launch_bounds__(256) void k_edge(const int* __restrict__ ei,
                                              const float* __restrict__ dis,
                                              const float* __restrict__ h,
                                              float* __restrict__ agg) {
  int wave = (int)(((size_t)blockIdx.x * blockDim.x + threadIdx.x) >> 5);
  int lane = threadIdx.x & 31;
  if (wave >= N_EDGES) return;
  int src = ei[wave];
  int dst = ei[N_EDGES + wave];
  float norm = dis[src] * dis[dst];
  const float4 hv = *(const float4*)(h + (size_t)src * HID + lane * 4);
  float* a = agg + (size_t)dst * HID + lane * 4;
  atomicAdd(a + 0, hv.x * norm);
  atomicAdd(a + 1, hv.y * norm);
  atomicAdd(a + 2, hv.z * norm);
  atomicAdd(a + 3, hv.w * norm);
}

// ---------------- bias + relu (in place) ----------------
__global__ void k_bias_relu(float* __restrict__ agg, const float* __restrict__ b) {
  size_t i = (size_t)blockIdx.x * blockDim.x + threadIdx.x;
  int f = (int)(i & (HID - 1));
  float v = agg[i] + b[f];
  agg[i] = v > 0.f ? v : 0.f;
}

// ---------------- pooling ----------------
__global__ void k_zero(float* __restrict__ p, int n) {
  int i = blockIdx.x * blockDim.x + threadIdx.x;
  if (i < n) p[i] = 0.f;
}

__global__ void k_pool(const float* __restrict__ h, const int* __restrict__ batch,
                       float* __restrict__ sum, int* __restrict__ mx,
                       float* __restrict__ cnt) {
  size_t i = (size_t)blockIdx.x * blockDim.x + threadIdx.x;   // N*HID
  int n = (int)(i >> 7), f = (int)(i & (HID - 1));
  int g = batch[n];
  float v = h[i];
  atomicAdd(&sum[g * HID + f], v);
  atomicMax(&mx[g * HID + f], __float_as_int(v));  // v >= 0 post-relu: int order == float order
  if (f == 0) atomicAdd(&cnt[g], 1.0f);
}

__global__ void k_feats(const float* __restrict__ sum, const int* __restrict__ mx,
                        const float* __restrict__ cnt, const float* __restrict__ gs,
                        float* __restrict__ feats) {
  int i = blockIdx.x * blockDim.x + threadIdx.x;   // G*HID
  int g = i >> 7, f = i & (HID - 1);
  float c = cnt[g]; c = c > 1.f ? c : 1.f;
  float s = sum[g * HID + f];
  feats[g * FEAT + f]           = s / c;                       // mean
  feats[g * FEAT + HID + f]     = __int_as_float(mx[g * HID + f]); // max
  feats[g * FEAT + 2 * HID + f] = s;                           // sum
  if (f < STATS) feats[g * FEAT + 3 * HID + f] = gs[g * STATS + f];
}

// ---------------- tiny MLP head ----------------
__global__ void k_mlp1(const float* __restrict__ feats, const float* __restrict__ Wm1,
                       const float* __restrict__ bm1, float* __restrict__ hid) {
  int i = blockIdx.x * blockDim.x + threadIdx.x;   // G*MLP_HID = 4096
  int g = i >> 6, j = i & (MLP_HID - 1);
  float acc = bm1[j];
  for (int k = 0; k < FEAT; ++k) acc += feats[g * FEAT + k] * Wm1[k * MLP_HID + j];
  hid[i] = acc > 0.f ? acc : 0.f;
}

__global__ void k_mlp2(const float* __restrict__ hid, const float* __restrict__ Wm2,
                       const float* __restrict__ bm2, float* __restrict__ out) {
  int g = blockIdx.x * blockDim.x + threadIdx.x;
  if (g < N_GRAPH) {
    float acc = bm2[0];
    for (int j = 0; j < MLP_HID; ++j) acc += hid[g * MLP_HID + j] * Wm2[j];
    out[g] = acc;
  }
}

extern "C" void kernel_launch(void* const* d_in, const int* in_sizes, int n_in,
                              void* d_out, int out_size, void* d_ws, size_t ws_size,
                              hipStream_t stream) {
  const float* x   = (const float*)d_in[0];
  const int*   ei  = (const int*)  d_in[1];   // [2, E] flat
  const int*   bat = (const int*)  d_in[2];
  const float* gs  = (const float*)d_in[3];
  const float* W1  = (const float*)d_in[4];
  const float* b1  = (const float*)d_in[5];
  const float* W2  = (const float*)d_in[6];
  const float* b2  = (const float*)d_in[7];
  const float* W3  = (const float*)d_in[8];
  const float* b3  = (const float*)d_in[9];
  const float* Wm1 = (const float*)d_in[10];
  const float* bm1 = (const float*)d_in[11];
  const float* Wm2 = (const float*)d_in[12];
  const float* bm2 = (const float*)d_in[13];

  float* ws    = (float*)d_ws;
  float* bufA  = ws;                        // [N, HID] GEMM output
  float* bufB  = bufA + (size_t)N_NODES * HID;  // [N, HID] aggregation / layer input
  float* dis   = bufB + (size_t)N_NODES * HID;  // [N]
  float* sum   = dis + N_NODES;             // [G, HID]
  int*   mx    = (int*)(sum + N_GRAPH * HID);   // [G, HID] (bit-punned fp32)
  float* cnt   = (float*)mx + N_GRAPH * HID;    // [G]
  float* feats = cnt + N_GRAPH;             // [G, FEAT]
  float* hid   = feats + N_GRAPH * FEAT;    // [G, MLP_HID]

  const int TB = 256;
  const int gridN   = (N_NODES + TB - 1) / TB;            // 196
  const int gridE   = (N_EDGES + TB - 1) / TB;            // 3125
  const int gridNH  = (N_NODES * HID) / TB;               // 25000
  const int gridEW  = (N_EDGES * 32) / TB;                // 100000 (wave/edge)
  const int poolElems = 2 * N_GRAPH * HID + N_GRAPH;      // sum + mx + cnt (contiguous)

  // norm precompute
  k_zero  <<<(poolElems + TB - 1) / TB, TB, 0, stream>>>(sum, poolElems);
  k_fill1 <<<gridN, TB, 0, stream>>>(dis);
  k_deg   <<<gridE, TB, 0, stream>>>(ei, dis);
  k_rsqrt <<<gridN, TB, 0, stream>>>(dis);

  // layer 1
  k_gemm_wmma <<<N_NODES / 16, TB, 0, stream>>>(x, W1, bufA, IN_DIM);
  k_selfinit  <<<gridNH, TB, 0, stream>>>(bufA, dis, bufB);
  k_edge      <<<gridEW, TB, 0, stream>>>(ei, dis, bufA, bufB);
  k_bias_relu <<<gridNH, TB, 0, stream>>>(bufB, b1);

  // layer 2
  k_gemm_wmma <<<N_NODES / 16, TB, 0, stream>>>(bufB, W2, bufA, HID);
  k_selfinit  <<<gridNH, TB, 0, stream>>>(bufA, dis, bufB);
  k_edge      <<<gridEW, TB, 0, stream>>>(ei, dis, bufA, bufB);
  k_bias_relu <<<gridNH, TB, 0, stream>>>(bufB, b2);

  // layer 3
  k_gemm_wmma <<<N_NODES / 16, TB, 0, stream>>>(bufB, W3, bufA, HID);
  k_selfinit  <<<gridNH, TB, 0, stream>>>(bufA, dis, bufB);
  k_edge      <<<gridEW, TB, 0, stream>>>(ei, dis, bufA, bufB);
  k_bias_relu <<<gridNH, TB, 0, stream>>>(bufB, b3);

  // readout + head
  k_pool  <<<gridNH, TB, 0, stream>>>(bufB, bat, sum, mx, cnt);
  k_feats <<<(N_GRAPH * HID) / TB, TB, 0, stream>>>(sum, mx, cnt, gs, feats);
  k_mlp1  <<<(N_GRAPH * MLP_HID) / TB, TB, 0, stream>>>(feats, Wm1, bm1, hid);
  k_mlp2  <<<1, 64, 0, stream>>>(hid, Wm2, bm2, (float*)d_out);
}